// S4Regressor_85813446574303
// MI455X (gfx1250) — compile-verified
//
#include <hip/hip_runtime.h>
#include <hip/hip_bf16.h>
#include <math.h>

// Problem constants (from reference)
#define Bv 32
#define Lv 4096
#define Cv 128
#define Nv 64
#define DEPTHv 4
#define Tc 64            // chunk length
#define NCH (Lv / Tc)    // 64 chunks
#define SN 128           // state rows = 2N (re/im interleaved)
#define WOUT_LDS_STRIDE 130   // padded row stride to dodge bank conflicts

typedef float v2f __attribute__((ext_vector_type(2)));
typedef float v8f __attribute__((ext_vector_type(8)));

// D = A(16x4,f32) * B(4x16,f32) + C(16x16,f32)   -> v_wmma_f32_16x16x4_f32
__device__ __forceinline__ v8f wmma_f32(v2f a, v2f b, v8f c) {
  return __builtin_amdgcn_wmma_f32_16x16x4_f32(
      /*neg_a=*/false, a, /*neg_b=*/false, b,
      /*c_mod=*/(short)0, c, /*reuse_a=*/false, /*reuse_b=*/false);
}

// ---- CDNA5 async global->LDS copy (ASYNCcnt path) --------------------------
#if defined(__has_builtin)
#if __has_builtin(__builtin_amdgcn_global_load_async_to_lds_b32)
#define HAVE_ASYNC_LDS_BUILTIN 1
#endif
#if __has_builtin(__builtin_amdgcn_s_wait_asynccnt)
#define HAVE_WAIT_ASYNC_BUILTIN 1
#endif
#endif

typedef __attribute__((address_space(1))) int g_int;
typedef __attribute__((address_space(3))) int l_int;

__device__ __forceinline__ void async_load_f32(const float* g, float* l) {
#ifdef HAVE_ASYNC_LDS_BUILTIN
  __builtin_amdgcn_global_load_async_to_lds_b32(
      (g_int*)(void*)g, (l_int*)(void*)l, 0, 0);
#else
  __attribute__((address_space(3))) float* lp =
      (__attribute__((address_space(3))) float*)l;
  asm volatile("global_load_async_to_lds_b32 %0, %1, off"
               :: "v"(lp), "v"(g) : "memory");
#endif
}

__device__ __forceinline__ void wait_async0() {
#ifdef HAVE_WAIT_ASYNC_BUILTIN
  __builtin_amdgcn_s_wait_asynccnt(0);
#else
  asm volatile("s_wait_asynccnt 0x0" ::: "memory");
#endif
}

// ---------------------------------------------------------------------------
// Per-layer, per-channel S4D matrices:
//   Kvec[l]      = 2*Re sum_n Cbar_n lam_n^l           (Toeplitz kernel, l<Tc)
//   Wout[l][2n]  = 2*Re(Cbar_n lam_n^l), [2n+1] = -2*Im(...)
//   Win [2n][t]  = Re(lam_n^(Tc-t)),     [2n+1][t] = Im(...)
//   LamT[2n/2n+1]= Re/Im(lam_n^Tc)
// with lam = exp(dt*A), Cbar = (C_re+iC_im)*(exp(dt*A)-1)/A
// ---------------------------------------------------------------------------
__global__ __launch_bounds__(256)
void s4d_precompute(const float* __restrict__ log_dt,
                    const float* __restrict__ A_real,
                    const float* __restrict__ A_imag,
                    const float* __restrict__ C_re,
                    const float* __restrict__ C_im,
                    int d,
                    float* __restrict__ gKvec,
                    float* __restrict__ gWout,
                    float* __restrict__ gWin,
                    float* __restrict__ gLamT) {
  const int c = blockIdx.x;
  const int tid = threadIdx.x;
  __shared__ float sAr[Nv], sAi[Nv], sCr[Nv], sCi[Nv];

  const float dt = expf(log_dt[d * Cv + c]);
  if (tid < Nv) {
    const int idx = (d * Cv + c) * Nv + tid;
    const float are = -expf(A_real[idx]);   // Re(A)
    const float aim = A_imag[idx];          // Im(A)
    const float dr = are * dt, di = aim * dt;   // dt*A
    // E = exp(dt*A) - 1
    const float eg = expf(dr);
    const float er = eg * cosf(di) - 1.0f;
    const float ei = eg * sinf(di);
    // F = E / A
    const float den = are * are + aim * aim;
    const float fr = (er * are + ei * aim) / den;
    const float fi = (ei * are - er * aim) / den;
    const float cr = C_re[idx], ci = C_im[idx];
    sAr[tid] = dr;
    sAi[tid] = di;
    sCr[tid] = cr * fr - ci * fi;           // Re(Cbar)
    sCi[tid] = cr * fi + ci * fr;           // Im(Cbar)
    // lam^Tc
    const float g = expf(dr * (float)Tc);
    gLamT[c * SN + 2 * tid]     = g * cosf(di * (float)Tc);
    gLamT[c * SN + 2 * tid + 1] = g * sinf(di * (float)Tc);
  }
  __syncthreads();

  if (tid < Tc) {
    const float l = (float)tid;
    float acc = 0.0f;
    for (int n = 0; n < Nv; ++n) {
      const float g  = expf(sAr[n] * l);
      const float lr = g * cosf(sAi[n] * l);
      const float li = g * sinf(sAi[n] * l);
      acc += sCr[n] * lr - sCi[n] * li;
    }
    gKvec[c * Tc + tid] = 2.0f * acc;
  }

  for (int i = tid; i < Tc * SN; i += blockDim.x) {
    const int l = i >> 7;            // / SN
    const int j = i & (SN - 1);
    const int n = j >> 1;
    const float g  = expf(sAr[n] * (float)l);
    const float lr = g * cosf(sAi[n] * (float)l);
    const float li = g * sinf(sAi[n] * (float)l);
    gWout[(size_t)c * Tc * SN + i] =
        (j & 1) ? -2.0f * (sCr[n] * li + sCi[n] * lr)
                :  2.0f * (sCr[n] * lr - sCi[n] * li);
  }

  for (int i = tid; i < SN * Tc; i += blockDim.x) {
    const int j = i >> 6;            // / Tc
    const int t = i & (Tc - 1);
    const int n = j >> 1;
    const float p = (float)(Tc - t); // powers Tc .. 1
    const float g = expf(sAr[n] * p);
    gWin[(size_t)c * SN * Tc + i] = (j & 1) ? g * sinf(sAi[n] * p)
                                            : g * cosf(sAi[n] * p);
  }
}

// ---------------------------------------------------------------------------
// Chunked SSM scan.  One block per channel; 8 waves; 64 sequential chunks.
// Per chunk:  Y(64x32) = Ktri(64x64)@X(64x32) + Wout(64x128)@S(128x32) + Dskip*X
//             S(128x32) <- lam^Tc (.) S + Win(128x64)@X
// X chunks are double-buffered via async global->LDS copies (ASYNCcnt).
// Dynamic LDS layout (floats):
//   sWout: Tc*130 = 8320 | sKv: 64 | sLamT: 128 | sS: SN*Bv = 4096
//   sX0: Tc*Bv = 2048    | sX1: 2048              -> 16704 f = 66816 B
// ---------------------------------------------------------------------------
#define SCAN_LDS_FLOATS (Tc * WOUT_LDS_STRIDE + Tc + SN + SN * Bv + 2 * Tc * Bv)
#define SCAN_LDS_BYTES  (SCAN_LDS_FLOATS * 4)

__global__ __launch_bounds__(256)
void s4d_chunk_scan(const float* __restrict__ xin,    // [B,L,C]
                    const float* __restrict__ gKvec,
                    const float* __restrict__ gWout,
                    const float* __restrict__ gWin,
                    const float* __restrict__ gLamT,
                    const float* __restrict__ Dskip_d, // [C] for this layer
                    float* __restrict__ ybuf) {        // [B,L,C]
  const int c    = blockIdx.x;
  const int tid  = threadIdx.x;
  const int lane = tid & 31;    // wave32
  const int wave = tid >> 5;    // 0..7
  const int hi   = lane >> 4;   // 0/1
  const int lo   = lane & 15;

  extern __shared__ float smem[];
  float* sWout = smem;                              // Tc*130
  float* sKv   = sWout + Tc * WOUT_LDS_STRIDE;      // Tc
  float* sLamT = sKv + Tc;                          // SN
  float* sS    = sLamT + SN;                        // SN*Bv
  float* sX0   = sS + SN * Bv;                      // Tc*Bv
  float* sX1   = sX0 + Tc * Bv;                     // Tc*Bv

  // per-channel constants: Wout via async copy, rest via normal path
  for (int i = tid; i < Tc * SN; i += 256) {
    const int l = i >> 7, j = i & (SN - 1);
    async_load_f32(&gWout[(size_t)c * Tc * SN + i], &sWout[l * WOUT_LDS_STRIDE + j]);
  }
  if (tid < Tc) sKv[tid]   = gKvec[c * Tc + tid];
  if (tid < SN) sLamT[tid] = gLamT[c * SN + tid];
  for (int i = tid; i < SN * Bv; i += 256) sS[i] = 0.0f;

  // kick off async copy of chunk 0 into sX0
  for (int i = tid; i < Tc * Bv; i += 256) {
    const int t = i >> 5, b = i & 31;
    async_load_f32(&xin[((size_t)b * Lv + t) * Cv + c], &sX0[i]);
  }

  const float dsk = Dskip_d[c];
  const float* winC = gWin + (size_t)c * SN * Tc;

  // Hoist Win A-fragments (constant across chunks); wave owns rows j=16w..16w+15
  v2f afW[16];
  {
    const int jg = wave * 16 + lo;
#pragma unroll
    for (int s = 0; s < 16; ++s) {
      const int ka = s * 4 + 2 * hi;
      v2f a;
      a.x = winC[jg * Tc + ka];
      a.y = winC[jg * Tc + ka + 1];
      afW[s] = a;
    }
  }

  wait_async0();       // our async writes (Wout + X0) done
  __syncthreads();     // everyone's done

  const int mi = wave >> 1;        // Y tile row (0..3)
  const int ni = wave & 1;         // Y tile col (0..1)
  const int lg = mi * 16 + lo;     // output row (chunk-local l) for A frags

  // Hoist chunk-invariant Toeplitz A-fragments (triangular, from sKv)
  v2f afK[16];
#pragma unroll
  for (int s = 0; s < 16; ++s) {
    const int ka = s * 4 + 2 * hi;
    v2f a;
    a.x = (ka     <= lg) ? sKv[lg - ka]     : 0.0f;
    a.y = (ka + 1 <= lg) ? sKv[lg - ka - 1] : 0.0f;
    afK[s] = a;
  }

  for (int k = 0; k < NCH; ++k) {
    float* sXc = (k & 1) ? sX1 : sX0;
    float* sXn = (k & 1) ? sX0 : sX1;

    // overlap: async-copy next chunk's X while we compute this one
    if (k + 1 < NCH) {
      for (int i = tid; i < Tc * Bv; i += 256) {
        const int t = i >> 5, b = i & 31;
        async_load_f32(&xin[((size_t)b * Lv + ((k + 1) * Tc + t)) * Cv + c], &sXn[i]);
      }
    }

    // ---- Y = Ktri @ X + Wout @ S ----
    v8f acc = {};
#pragma unroll
    for (int s = 0; s < 16; ++s) {          // Toeplitz part, K=64
      const int ka = s * 4 + 2 * hi;
      v2f bf;
      bf.x = sXc[ka * Bv + ni * 16 + lo];
      bf.y = sXc[(ka + 1) * Bv + ni * 16 + lo];
      acc = wmma_f32(afK[s], bf, acc);
    }
#pragma unroll
    for (int s = 0; s < 32; ++s) {          // state-output part, K=128
      const int ka = s * 4 + 2 * hi;
      v2f a, bf;
      a.x = sWout[lg * WOUT_LDS_STRIDE + ka];
      a.y = sWout[lg * WOUT_LDS_STRIDE + ka + 1];
      bf.x = sS[ka * Bv + ni * 16 + lo];
      bf.y = sS[(ka + 1) * Bv + ni * 16 + lo];
      acc = wmma_f32(a, bf, acc);
    }
    // write Y + D-skip  (C/D layout: VGPR r -> M=r+8*hi, N=lo)
#pragma unroll
    for (int r = 0; r < 8; ++r) {
      const int m  = r + 8 * hi;
      const int tl = mi * 16 + m;          // chunk-local position
      const int b  = ni * 16 + lo;         // batch
      const float xv = sXc[tl * Bv + b];
      ybuf[((size_t)b * Lv + (k * Tc + tl)) * Cv + c] = acc[r] + dsk * xv;
    }
    __syncthreads();  // everyone done reading sS/sX before state update

    // ---- U = Win @ X ;  S <- lam^Tc (.) S + U ----
#pragma unroll
    for (int n2 = 0; n2 < 2; ++n2) {
      v8f u = {};
#pragma unroll
      for (int s = 0; s < 16; ++s) {
        const int ka = s * 4 + 2 * hi;
        v2f bf;
        bf.x = sXc[ka * Bv + n2 * 16 + lo];
        bf.y = sXc[(ka + 1) * Bv + n2 * 16 + lo];
        u = wmma_f32(afW[s], bf, u);
      }
      const int jb = wave * 16 + 8 * hi;   // even base -> (re,im) pairs in-lane
      const int b  = n2 * 16 + lo;
#pragma unroll
      for (int p = 0; p < 4; ++p) {
        const int j0 = jb + 2 * p;
        const float ltr = sLamT[j0], lti = sLamT[j0 + 1];
        const float sr = sS[j0 * Bv + b], si = sS[(j0 + 1) * Bv + b];
        sS[j0 * Bv + b]       = ltr * sr - lti * si + u[2 * p];
        sS[(j0 + 1) * Bv + b] = ltr * si + lti * sr + u[2 * p + 1];
      }
    }
    wait_async0();    // our async writes into sXn have landed
    __syncthreads();  // everyone's writes visible; state update complete
  }
}

// ---------------------------------------------------------------------------
// x' = LayerNorm(y + residual) over C, per (b,l) row
// ---------------------------------------------------------------------------
__global__ __launch_bounds__(Cv)
void layernorm_add(const float* __restrict__ y, const float* __restrict__ res,
                   const float* __restrict__ w, const float* __restrict__ b,
                   float* __restrict__ out) {
  const int row = blockIdx.x;     // b*L + l
  const int c   = threadIdx.x;    // 0..127
  __shared__ float red[Cv];
  const float v = y[(size_t)row * Cv + c] + res[(size_t)row * Cv + c];
  red[c] = v;
  __syncthreads();
  for (int s = Cv / 2; s > 0; s >>= 1) {
    if (c < s) red[c] += red[c + s];
    __syncthreads();
  }
  const float mu = red[0] * (1.0f / Cv);
  __syncthreads();
  const float dv = v - mu;
  red[c] = dv * dv;
  __syncthreads();
  for (int s = Cv / 2; s > 0; s >>= 1) {
    if (c < s) red[c] += red[c + s];
    __syncthreads();
  }
  const float var = red[0] * (1.0f / Cv);
  out[(size_t)row * Cv + c] = dv * rsqrtf(var + 1e-5f) * w[c] + b[c];
}

// ---------------------------------------------------------------------------
// out[b,l] = x[b,l,:] . head_w + head_b
// ---------------------------------------------------------------------------
__global__ __launch_bounds__(Cv)
void head_proj(const float* __restrict__ x, const float* __restrict__ hw,
               const float* __restrict__ hb, float* __restrict__ out) {
  const int row = blockIdx.x;
  const int c   = threadIdx.x;
  __shared__ float red[Cv];
  red[c] = x[(size_t)row * Cv + c] * hw[c];
  __syncthreads();
  for (int s = Cv / 2; s > 0; s >>= 1) {
    if (c < s) red[c] += red[c + s];
    __syncthreads();
  }
  if (c == 0) out[row] = red[0] + hb[0];
}

// ---------------------------------------------------------------------------
extern "C" void kernel_launch(void* const* d_in, const int* in_sizes, int n_in,
                              void* d_out, int out_size, void* d_ws, size_t ws_size,
                              hipStream_t stream) {
  (void)in_sizes; (void)n_in; (void)out_size; (void)ws_size;
  const float* x      = (const float*)d_in[0];
  const float* log_dt = (const float*)d_in[1];
  const float* A_real = (const float*)d_in[2];
  const float* A_imag = (const float*)d_in[3];
  const float* C_re   = (const float*)d_in[4];
  const float* C_im   = (const float*)d_in[5];
  const float* Dskip  = (const float*)d_in[6];
  const float* norm_w = (const float*)d_in[7];
  const float* norm_b = (const float*)d_in[8];
  const float* head_w = (const float*)d_in[9];
  const float* head_b = (const float*)d_in[10];
  float* out = (float*)d_out;

  float* ws = (float*)d_ws;
  const size_t act = (size_t)Bv * Lv * Cv;       // 16.78M floats
  float* ybuf  = ws;                              // [B,L,C]
  float* xbuf  = ws + act;                        // [B,L,C]
  float* gKvec = ws + 2 * act;                    // C*Tc
  float* gWout = gKvec + (size_t)Cv * Tc;         // C*Tc*SN
  float* gWin  = gWout + (size_t)Cv * Tc * SN;    // C*SN*Tc
  float* gLamT = gWin  + (size_t)Cv * SN * Tc;    // C*SN

  // allow > default dynamic LDS (66816 B; WGP has 320 KB) — host-side, capture-safe
  (void)hipFuncSetAttribute((const void*)s4d_chunk_scan,
                            hipFuncAttributeMaxDynamicSharedMemorySize,
                            SCAN_LDS_BYTES);

  for (int d = 0; d < DEPTHv; ++d) {
    const float* xin = (d == 0) ? x : xbuf;
    s4d_precompute<<<Cv, 256, 0, stream>>>(log_dt, A_real, A_imag, C_re, C_im, d,
                                           gKvec, gWout, gWin, gLamT);
    s4d_chunk_scan<<<Cv, 256, SCAN_LDS_BYTES, stream>>>(xin, gKvec, gWout, gWin,
                                                        gLamT, Dskip + d * Cv, ybuf);
    layernorm_add<<<Bv * Lv, Cv, 0, stream>>>(ybuf, xin, norm_w + d * Cv,
                                              norm_b + d * Cv, xbuf);
  }
  head_proj<<<Bv * Lv, Cv, 0, stream>>>(xbuf, head_w, head_b, out);
}